// LatticeAttentionHead_25400436589157
// MI455X (gfx1250) — compile-verified
//
#include <hip/hip_runtime.h>
#include <math.h>

// ---------------------------------------------------------------------------
// Problem constants (match reference)
// ---------------------------------------------------------------------------
#define BATCH    4
#define SEQ      2048
#define DMODEL   1024
#define DK       64
#define NNEIGH   4

typedef float v2f __attribute__((ext_vector_type(2)));
typedef float v8f __attribute__((ext_vector_type(8)));

// D = A(16x4,f32) * B(4x16,f32) + C(16x16,f32)  -- CDNA5 wave32 WMMA
__device__ __forceinline__ v8f wmma_f32(v2f a, v2f b, v8f c) {
    return __builtin_amdgcn_wmma_f32_16x16x4_f32(
        /*neg_a=*/false, a, /*neg_b=*/false, b,
        /*c_mod=*/(short)0, c, /*reuse_a=*/false, /*reuse_b=*/false);
}

// CDNA5 async global->LDS copy (ASYNCcnt-tracked, no VGPR data path).
// LDS destination is the low 32 bits of the generic pointer (flat LDS
// aperture keeps the byte offset in addr[31:0]).
__device__ __forceinline__ void async_ld_b128(const void* lds_dst, const void* gsrc) {
    const uint32_t loff  = (uint32_t)(uintptr_t)lds_dst;
    const uint64_t gaddr = (uint64_t)(uintptr_t)gsrc;
    asm volatile("global_load_async_to_lds_b128 %0, %1, off"
                 :: "v"(loff), "v"(gaddr)
                 : "memory");
}
__device__ __forceinline__ void wait_async0() {
    asm volatile("s_wait_asynccnt 0" ::: "memory");
}

// ---------------------------------------------------------------------------
// Kernel A: scalar lattice phase update -> lattice_influence (one float)
// ---------------------------------------------------------------------------
__global__ void lattice_phase_kernel(const float* __restrict__ nb_phase,
                                     const float* __restrict__ nb_dist,
                                     const float* __restrict__ freq,
                                     const float* __restrict__ phase,
                                     const float* __restrict__ coupling,
                                     const float* __restrict__ lat_const,
                                     float* __restrict__ out_influence) {
    if (threadIdx.x == 0 && blockIdx.x == 0) {
        const float ph = phase[0];
        const float a  = lat_const[0];
        float cf = 0.0f;
        #pragma unroll
        for (int i = 0; i < NNEIGH; ++i)
            cf += expf(-nb_dist[i] / a) * sinf(nb_phase[i] - ph);
        const float pc   = coupling[0] * cf;
        const float newp = ph + 0.01f * (freq[0] + pc);
        out_influence[0] = cosf(newp) * 0.1f;
    }
}

// ---------------------------------------------------------------------------
// Kernel B: projection GEMM  Y[M,64] = X[M,1024] * W[1024,64]
//   block = 128 threads (4 waves), M-block = 64 rows, full N = 64.
//   K staged in 32-wide chunks via double-buffered async global->LDS DMA:
//   chunk k+1 streams into the alternate buffer while WMMAs consume chunk k.
// ---------------------------------------------------------------------------
#define KC 32

__global__ __launch_bounds__(128)
void proj_kernel(const float* __restrict__ X, const float* __restrict__ W,
                 float* __restrict__ Y) {
    __shared__ float As[2][64][KC + 4];   // +4: 16B-aligned rows, fewer conflicts
    __shared__ float Bs[2][KC][DK];

    const int tid   = threadIdx.x;
    const int lane  = tid & 31;
    const int wave  = tid >> 5;
    const int m0    = blockIdx.x * 64;
    const int r15   = lane & 15;
    const int khalf = (lane >> 4) * 2;

    v8f acc[4] = {};

    // issue async stage of one K-chunk into buffer `buf`
    auto stage = [&](int buf, int k0) {
        for (int i = tid; i < 64 * (KC / 4); i += 128) {     // 4 b128 per thread
            const int row = i / (KC / 4);
            const int c4  = i % (KC / 4);
            async_ld_b128(&As[buf][row][c4 * 4],
                          X + (size_t)(m0 + row) * DMODEL + k0 + c4 * 4);
        }
        for (int i = tid; i < KC * (DK / 4); i += 128) {     // 4 b128 per thread
            const int row = i / (DK / 4);
            const int c4  = i % (DK / 4);
            async_ld_b128(&Bs[buf][row][c4 * 4],
                          W + (size_t)(k0 + row) * DK + c4 * 4);
        }
    };

    stage(0, 0);
    wait_async0();
    __syncthreads();

    int buf = 0;
    for (int k0 = 0; k0 < DMODEL; k0 += KC) {
        if (k0 + KC < DMODEL) stage(buf ^ 1, k0 + KC);   // overlap DMA w/ WMMA

        const int arow = wave * 16 + r15;
        #pragma unroll
        for (int kk = 0; kk < KC; kk += 4) {
            v2f a;
            a.x = As[buf][arow][kk + khalf];
            a.y = As[buf][arow][kk + khalf + 1];
            #pragma unroll
            for (int n = 0; n < 4; ++n) {
                v2f b;
                const int col = n * 16 + r15;
                b.x = Bs[buf][kk + khalf][col];
                b.y = Bs[buf][kk + khalf + 1][col];
                acc[n] = wmma_f32(a, b, acc[n]);
            }
        }

        wait_async0();      // own wave's prefetch done
        __syncthreads();    // everyone's done reading old buf + staging new
        buf ^= 1;
    }

    // Write 16x64 tile. C layout: VGPR v -> row v (lanes 0-15) / v+8 (lanes 16-31)
    const int rbase = (lane >> 4) * 8;
    #pragma unroll
    for (int n = 0; n < 4; ++n)
        #pragma unroll
        for (int v = 0; v < 8; ++v) {
            const int row = m0 + wave * 16 + rbase + v;
            Y[(size_t)row * DK + n * 16 + r15] = acc[n][v];
        }
}

// ---------------------------------------------------------------------------
// Kernel C: attention for one (batch, 16-row query block) per 32-thread block.
//   Full 16 x 2048 score row-tile in LDS (~128 KB; CDNA5 WGP has 320 KB).
//   K tiles are double-buffered through LDS via async DMA so the next tile
//   streams in while the current tile's 16-WMMA chain executes.
// ---------------------------------------------------------------------------
__global__ __launch_bounds__(32)
void attn_kernel(const float* __restrict__ Qm, const float* __restrict__ Km,
                 const float* __restrict__ Vm, const int* __restrict__ mask,
                 const float* __restrict__ infl_p,
                 float* __restrict__ outO, float* __restrict__ outW) {
    __shared__ float sc[16][SEQ + 4];
    __shared__ float Ks[2][16][DK];

    const int lane  = threadIdx.x;
    const int b     = blockIdx.y;
    const int q0    = blockIdx.x * 16;
    const int r15   = lane & 15;
    const int hi    = lane >> 4;       // 0 or 1
    const int khalf = hi * 2;
    const size_t base = (size_t)b * SEQ * DK;

    const float infl  = *infl_p;
    const float scale = 0.125f;        // 1/sqrt(64)

    // Preload the 16 A-fragments of this Q tile (K-dim = 64 -> 16 k-steps)
    v2f qa[16];
    #pragma unroll
    for (int kk = 0; kk < 16; ++kk) {
        const float* p = Qm + base + (size_t)(q0 + r15) * DK + kk * 4 + khalf;
        qa[kk].x = p[0];
        qa[kk].y = p[1];
    }

    // async stage of one 16x64 K tile (4 KB) -- 8 coalesced b128 per lane
    auto stageK = [&](int buf, int j) {
        for (int i = lane; i < 16 * (DK / 4); i += 32) {
            const int row = i / (DK / 4);
            const int c4  = i % (DK / 4);
            async_ld_b128(&Ks[buf][row][c4 * 4],
                          Km + base + (size_t)(j * 16 + row) * DK + c4 * 4);
        }
    };

    // ---- scores: iterate over all 128 key tiles (double-buffered) ----
    stageK(0, 0);
    wait_async0();
    int buf = 0;
    for (int j = 0; j < SEQ / 16; ++j) {
        if (j + 1 < SEQ / 16) stageK(buf ^ 1, j + 1);   // overlap DMA w/ WMMA

        v8f acc = {};
        #pragma unroll
        for (int kk = 0; kk < 16; ++kk) {
            // B frag: B[k][n] = K[j*16+n][k]; contiguous b64 from LDS
            v2f bf;
            bf.x = Ks[buf][r15][kk * 4 + khalf];
            bf.y = Ks[buf][r15][kk * 4 + khalf + 1];
            acc = wmma_f32(qa[kk], bf, acc);
        }
        #pragma unroll
        for (int v = 0; v < 8; ++v) {
            const int row = hi * 8 + v;
            const int col = j * 16 + r15;
            const float s = acc[v] * scale + infl;
            const int  mv = mask[(size_t)(q0 + row) * SEQ + col];
            sc[row][col] = (mv == 0) ? -1e9f : s;
        }

        wait_async0();          // next tile landed (single wave: no barrier)
        buf ^= 1;
    }

    // ---- softmax per row + write attention weights ----
    for (int r = 0; r < 16; ++r) {
        float m = -3.402823466e38f;
        for (int c = lane; c < SEQ; c += 32) m = fmaxf(m, sc[r][c]);
        #pragma unroll
        for (int off = 16; off > 0; off >>= 1) m = fmaxf(m, __shfl_xor(m, off, 32));

        float sum = 0.0f;
        for (int c = lane; c < SEQ; c += 32) {
            const float e = __expf(sc[r][c] - m);
            sc[r][c] = e;
            sum += e;
        }
        #pragma unroll
        for (int off = 16; off > 0; off >>= 1) sum += __shfl_xor(sum, off, 32);

        const float rinv = 1.0f / sum;
        float* wrow = outW + ((size_t)b * SEQ + (q0 + r)) * SEQ;
        for (int c = lane; c < SEQ; c += 32) {
            const float p = sc[r][c] * rinv;
            sc[r][c] = p;
            wrow[c]  = p;
        }
    }

    // ---- output tile: P(16x2048) @ V(2048x64) ----
    v8f oacc[4] = {};
    for (int kb = 0; kb < SEQ; kb += 4) {
        const int k0 = kb + khalf;
        v2f a;
        a.x = sc[r15][k0];
        a.y = sc[r15][k0 + 1];
        const float* v0 = Vm + base + (size_t)k0 * DK;
        __builtin_prefetch(v0 + 4 * DK, 0, 0);
        #pragma unroll
        for (int n = 0; n < 4; ++n) {
            v2f bf;
            const int col = n * 16 + r15;
            bf.x = v0[col];
            bf.y = v0[DK + col];
            oacc[n] = wmma_f32(a, bf, oacc[n]);
        }
    }
    #pragma unroll
    for (int n = 0; n < 4; ++n)
        #pragma unroll
        for (int v = 0; v < 8; ++v) {
            const int row = q0 + hi * 8 + v;
            outO[base + (size_t)row * DK + n * 16 + r15] = oacc[n][v];
        }
}

// ---------------------------------------------------------------------------
// Host-side launcher
// ---------------------------------------------------------------------------
extern "C" void kernel_launch(void* const* d_in, const int* in_sizes, int n_in,
                              void* d_out, int out_size, void* d_ws, size_t ws_size,
                              hipStream_t stream) {
    const float* query   = (const float*)d_in[0];
    const float* key     = (const float*)d_in[1];
    const float* value   = (const float*)d_in[2];
    const int*   mask    = (const int*)  d_in[3];
    const float* nbph    = (const float*)d_in[4];
    const float* nbdist  = (const float*)d_in[5];
    const float* W_q     = (const float*)d_in[6];
    const float* W_k     = (const float*)d_in[7];
    const float* W_v     = (const float*)d_in[8];
    const float* ifreq   = (const float*)d_in[9];
    const float* phase   = (const float*)d_in[10];
    const float* coup    = (const float*)d_in[11];
    const float* latc    = (const float*)d_in[12];

    // Workspace layout: Q | K | V | influence scalar
    const size_t projElems = (size_t)BATCH * SEQ * DK;   // 524288 floats each
    float* Qm   = (float*)d_ws;
    float* Km   = Qm + projElems;
    float* Vm   = Km + projElems;
    float* infl = Vm + projElems;

    // Output layout: output [B,S,DK] then attention_weights [B,S,S]
    float* outO = (float*)d_out;
    float* outW = outO + projElems;

    lattice_phase_kernel<<<1, 32, 0, stream>>>(nbph, nbdist, ifreq, phase,
                                               coup, latc, infl);

    const int mBlocks = (BATCH * SEQ) / 64;   // 128
    proj_kernel<<<dim3(mBlocks), 128, 0, stream>>>(query, W_q, Qm);
    proj_kernel<<<dim3(mBlocks), 128, 0, stream>>>(key,   W_k, Km);
    proj_kernel<<<dim3(mBlocks), 128, 0, stream>>>(value, W_v, Vm);

    attn_kernel<<<dim3(SEQ / 16, BATCH), 32, 0, stream>>>(Qm, Km, Vm, mask,
                                                          infl, outO, outW);
}